// FilterDetection_69724499083414
// MI455X (gfx1250) — compile-verified
//
#include <hip/hip_runtime.h>
#include <stdint.h>

// ---------------------------------------------------------------------------
// FilterDetection for MI455X (gfx1250):
//   boxes [8,200000,6] f32, classification [8,200000,80] f32
//   -> out_boxes [8,2000,6], out_scores [8,2000], out_labels [8,2000] (int32)
// Bandwidth-bound: 512MB classification read dominates (~22us @ 23.3TB/s).
// CDNA5 path: TENSOR_LOAD_TO_LDS (TDM DMA, one instruction per 80KB tile,
// hardware LDS padding) + s_wait_tensorcnt.
// ---------------------------------------------------------------------------

#define NB 8
#define NN 200000
#define NC 80
#define K_DET 2000
#define THRESH 0.05f

#define TILE_ROWS 256
#define TILES_PER_BATCH ((NN + TILE_ROWS - 1) / TILE_ROWS)   // 782
// TDM pads 4 dwords after every 16 data dwords: 80 -> 100 dwords per row.
#define ROW_DW_PADDED 100

#define HIST_BLOCKS 98           // 98 * 2048 >= 200000
#define HIST_CHUNK 2048

typedef unsigned int u32x4 __attribute__((ext_vector_type(4)));
typedef unsigned int u32x8 __attribute__((ext_vector_type(8)));

struct SelState {
    unsigned long long prefix;   // accumulated radix prefix -> final exact threshold key
    unsigned long long mask;     // which high bits of prefix are decided
    unsigned k;                  // remaining rank within current prefix
    unsigned pad0;
    unsigned long long pad1;
};                               // 32 bytes

// low 32 bits of a flat shared-aperture address == LDS byte offset (ISA 10.2)
__device__ __forceinline__ unsigned lds_off(const void* p) {
    return (unsigned)(uintptr_t)p;
}

// ---------------------------------------------------------------------------
// Pass 1: TDM-stage a 256x80 f32 tile into LDS (padded rows), then per-thread
// max/argmax over 80 classes, emit 64b order keys.
// ---------------------------------------------------------------------------
__global__ __launch_bounds__(256) void score_kernel(const float* __restrict__ cls,
                                                    unsigned long long* __restrict__ keys) {
    __shared__ __align__(16) float tile[TILE_ROWS * ROW_DW_PADDED];   // 100KB

    const unsigned b = blockIdx.x / TILES_PER_BATCH;
    const unsigned t = blockIdx.x % TILES_PER_BATCH;
    const unsigned tileStart = t * TILE_ROWS;

    if (threadIdx.x < 32) {   // wave 0 issues the DMA (EXEC ignored by TDM)
        unsigned long long gaddr =
            (unsigned long long)(uintptr_t)(cls + ((size_t)b * NN + tileStart) * NC);
        const unsigned rows = NN - tileStart;    // OOB rows (tail tile) read as zero
        const unsigned ldsBase = lds_off(tile);

        // D# group 0 (ISA 8.3): count=1 | lds_addr | global_addr[56:0] | type=2
        u32x4 g0;
        g0[0] = 1u;                                        // count=1, user mode
        g0[1] = ldsBase;                                   // lds_addr
        g0[2] = (unsigned)gaddr;                           // global_addr[31:0]
        g0[3] = (unsigned)((gaddr >> 32) & 0x1FFFFFFull)   // global_addr[56:32]
              | 0x80000000u;                               // type=2 ("image")
        // D# group 1 (ISA 8.4)
        u32x8 g1;
        g1[0] = (2u << 16)      // data_size = 4B
              | (1u << 20)      // pad_enable
              | (3u << 22)      // pad_interval: 16 dwords
              | (3u << 25);     // pad_amount:   4 dwords
        g1[1] = ((unsigned)NC & 0xFFFFu) << 16;            // tensor_dim0 lo (=80)
        g1[2] = ((rows & 0xFFFFu) << 16);                  // tensor_dim0 hi(0) | tensor_dim1 lo
        g1[3] = (rows >> 16) | ((unsigned)NC << 16);       // tensor_dim1 hi | tile_dim0=80
        g1[4] = (unsigned)TILE_ROWS;                       // tile_dim1=256, tile_dim2=0
        g1[5] = (unsigned)NC;                              // tensor_dim0_stride = 80 (lo)
        g1[6] = 0u;                                        // stride0 hi | stride1 lo
        g1[7] = 0u;                                        // stride1 hi (unused, 2D)

        asm volatile("tensor_load_to_lds %0, %1" :: "s"(g0), "s"(g1) : "memory");
        __builtin_amdgcn_s_wait_tensorcnt(0);
    }
    __syncthreads();

    const unsigned n = tileStart + threadIdx.x;
    if (n < NN) {
        const float* row = tile + (size_t)threadIdx.x * ROW_DW_PADDED;
        float best = -__builtin_inff();
        int bestc = 0;
        // 20 aligned float4 reads; padded layout: float4 j at dword (j/4)*20 + (j%4)*4
#pragma unroll
        for (int j = 0; j < 20; ++j) {
            const float4 v = *(const float4*)(row + ((j >> 2) * 20 + (j & 3) * 4));
            const int c0 = j * 4;
            if (v.x > best) { best = v.x; bestc = c0 + 0; }   // strict > keeps first max
            if (v.y > best) { best = v.y; bestc = c0 + 1; }
            if (v.z > best) { best = v.z; bestc = c0 + 2; }
            if (v.w > best) { best = v.w; bestc = c0 + 3; }
        }
        unsigned long long key = 0ull;
        if (best > THRESH) {
            unsigned bits = __float_as_uint(best);
            unsigned skey = (bits & 0x80000000u) ? ~bits : (bits | 0x80000000u); // order-preserving
            unsigned payload = (n << 7) | (unsigned)bestc;                        // n<18b, label<7b
            key = ((unsigned long long)skey << 32) | (unsigned long long)(0xFFFFFFFFu - payload);
        }
        keys[(size_t)b * NN + n] = key;   // descending key order == jax top_k order (stable ties)
    }
}

// ---------------------------------------------------------------------------
// Radix-select: 8-bit histogram pass restricted to the current prefix
// ---------------------------------------------------------------------------
__global__ __launch_bounds__(256) void hist_kernel(const unsigned long long* __restrict__ keys,
                                                   unsigned* __restrict__ hist,
                                                   const SelState* __restrict__ st, int shift) {
    __shared__ unsigned h[256];
    const unsigned b = blockIdx.x / HIST_BLOCKS;
    const unsigned c = blockIdx.x % HIST_BLOCKS;
    h[threadIdx.x] = 0u;
    __syncthreads();
    const unsigned long long mask = st[b].mask;
    const unsigned long long prefix = st[b].prefix;
    const unsigned long long* kb = keys + (size_t)b * NN;
    int start = (int)(c * HIST_CHUNK);
    int end = start + HIST_CHUNK; if (end > NN) end = NN;
    for (int i = start + (int)threadIdx.x; i < end; i += 256) {
        unsigned long long key = kb[i];
        if ((key & mask) == prefix)
            atomicAdd(&h[(unsigned)(key >> shift) & 255u], 1u);
    }
    __syncthreads();
    atomicAdd(&hist[b * 256u + threadIdx.x], h[threadIdx.x]);
}

__global__ __launch_bounds__(256) void select_kernel(unsigned* __restrict__ hist,
                                                     SelState* __restrict__ st, int shift) {
    const unsigned b = blockIdx.x;
    __shared__ unsigned h[256];
    h[threadIdx.x] = hist[b * 256u + threadIdx.x];
    __syncthreads();
    if (threadIdx.x == 0) {
        unsigned k = st[b].k;
        unsigned cum = 0;        // count strictly above chosen bin
        int v = 255;
        for (; v > 0; --v) {
            unsigned cnt = h[v];
            if (cum + cnt >= k) break;
            cum += cnt;
        }
        st[b].prefix |= ((unsigned long long)(unsigned)v) << shift;
        st[b].mask   |= (0xFFull << shift);
        st[b].k       = k - cum;
    }
    __syncthreads();
    hist[b * 256u + threadIdx.x] = 0u;   // re-zero for next pass
}

// ---------------------------------------------------------------------------
// Gather keys >= exact threshold T (unique keys -> exactly min(2000, #valid))
// ---------------------------------------------------------------------------
__global__ __launch_bounds__(256) void gather_kernel(const unsigned long long* __restrict__ keys,
                                                     const SelState* __restrict__ st,
                                                     unsigned long long* __restrict__ gathered,
                                                     unsigned* __restrict__ counts) {
    const unsigned b = blockIdx.x / TILES_PER_BATCH;
    const unsigned c = blockIdx.x % TILES_PER_BATCH;
    const unsigned i = c * 256u + threadIdx.x;
    if (i >= NN) return;
    const unsigned long long T = st[b].prefix;
    unsigned long long key = keys[(size_t)b * NN + i];
    if (key != 0ull && key >= T) {
        unsigned pos = atomicAdd(&counts[b], 1u);
        if (pos < 2048u) gathered[b * 2048u + pos] = key;
    }
}

// ---------------------------------------------------------------------------
// Bitonic sort 2048 keys in LDS, decode, gather boxes, write outputs
// ---------------------------------------------------------------------------
__global__ __launch_bounds__(1024) void output_kernel(const unsigned long long* __restrict__ gathered,
                                                      const float* __restrict__ boxes,
                                                      float* __restrict__ outBoxes,
                                                      float* __restrict__ outScores,
                                                      int* __restrict__ outLabels) {
    __shared__ unsigned long long s[2048];   // 16KB
    const unsigned b = blockIdx.x;
    const unsigned tid = threadIdx.x;
    s[tid]        = gathered[b * 2048u + tid];
    s[tid + 1024] = gathered[b * 2048u + tid + 1024];
    __syncthreads();

    // ascending bitonic sort; read reversed for descending ranks
    for (unsigned size = 2; size <= 2048; size <<= 1) {
        for (unsigned stride = size >> 1; stride >= 1; stride >>= 1) {
            unsigned i = 2u * stride * (tid / stride) + (tid % stride);
            unsigned j = i + stride;
            bool asc = ((i & size) == 0u);
            unsigned long long a = s[i], c = s[j];
            bool sw = asc ? (a > c) : (a < c);
            if (sw) { s[i] = c; s[j] = a; }
            __syncthreads();
        }
    }

    for (unsigned k = tid; k < K_DET; k += 1024) {
        unsigned long long key = s[2047 - k];      // rank k descending
        float* ob = outBoxes + ((size_t)b * K_DET + k) * 6;
        if (key != 0ull) {
            unsigned skey = (unsigned)(key >> 32);
            unsigned bits = (skey & 0x80000000u) ? (skey ^ 0x80000000u) : ~skey;
            unsigned payload = 0xFFFFFFFFu - (unsigned)key;
            unsigned n = payload >> 7;
            unsigned label = payload & 0x7Fu;
            const float* bp = boxes + ((size_t)b * NN + n) * 6;
#pragma unroll
            for (int j = 0; j < 6; ++j) ob[j] = bp[j];
            outScores[b * K_DET + k] = __uint_as_float(bits);
            outLabels[b * K_DET + k] = (int)label;
        } else {
#pragma unroll
            for (int j = 0; j < 6; ++j) ob[j] = -1.0f;
            outScores[b * K_DET + k] = -1.0f;
            outLabels[b * K_DET + k] = -1;
        }
    }
}

// ---------------------------------------------------------------------------
// Init workspace state each call (deterministic / graph-capture safe)
// ---------------------------------------------------------------------------
__global__ __launch_bounds__(1024) void init_kernel(unsigned* __restrict__ hist,
                                                    SelState* __restrict__ st,
                                                    unsigned long long* __restrict__ gathered,
                                                    unsigned* __restrict__ counts) {
    unsigned tid = blockIdx.x * blockDim.x + threadIdx.x;
    unsigned stride = gridDim.x * blockDim.x;
    for (unsigned i = tid; i < NB * 256u; i += stride) hist[i] = 0u;
    for (unsigned i = tid; i < NB * 2048u; i += stride) gathered[i] = 0ull;
    if (tid < NB) {
        st[tid].prefix = 0ull;
        st[tid].mask   = 0ull;
        st[tid].k      = K_DET;
        counts[tid]    = 0u;
    }
}

extern "C" void kernel_launch(void* const* d_in, const int* in_sizes, int n_in,
                              void* d_out, int out_size, void* d_ws, size_t ws_size,
                              hipStream_t stream) {
    (void)in_sizes; (void)n_in; (void)out_size; (void)ws_size;
    const float* boxes = (const float*)d_in[0];          // [8,200000,6]
    const float* cls   = (const float*)d_in[1];          // [8,200000,80]

    float* outBoxes  = (float*)d_out;                    // [8,2000,6]
    float* outScores = outBoxes + (size_t)NB * K_DET * 6;
    int*   outLabels = (int*)(outScores + (size_t)NB * K_DET);

    // workspace layout (~13MB)
    char* ws = (char*)d_ws;
    unsigned long long* keys = (unsigned long long*)ws;                       // 12.8MB
    size_t off = (size_t)NB * NN * sizeof(unsigned long long);
    unsigned* hist = (unsigned*)(ws + off);          off += NB * 256 * sizeof(unsigned);
    SelState* st   = (SelState*)(ws + off);          off += NB * sizeof(SelState);
    unsigned* counts = (unsigned*)(ws + off);        off += NB * sizeof(unsigned);
    off = (off + 255) & ~(size_t)255;
    unsigned long long* gathered = (unsigned long long*)(ws + off);           // 128KB

    hipLaunchKernelGGL(init_kernel, dim3(16), dim3(1024), 0, stream,
                       hist, st, gathered, counts);

    hipLaunchKernelGGL(score_kernel, dim3(NB * TILES_PER_BATCH), dim3(256), 0, stream,
                       cls, keys);

    for (int p = 0; p < 8; ++p) {
        int shift = 56 - p * 8;
        hipLaunchKernelGGL(hist_kernel, dim3(NB * HIST_BLOCKS), dim3(256), 0, stream,
                           keys, hist, st, shift);
        hipLaunchKernelGGL(select_kernel, dim3(NB), dim3(256), 0, stream,
                           hist, st, shift);
    }

    hipLaunchKernelGGL(gather_kernel, dim3(NB * TILES_PER_BATCH), dim3(256), 0, stream,
                       keys, st, gathered, counts);

    hipLaunchKernelGGL(output_kernel, dim3(NB), dim3(1024), 0, stream,
                       gathered, boxes, outBoxes, outScores, outLabels);
}